// LogicLayer_84000970375671
// MI455X (gfx1250) — compile-verified
//
#include <hip/hip_runtime.h>
#include <hip/hip_bf16.h>

// Problem constants (from the reference).
#define BATCH   2048
#define IN_DIM  512
#define NUM_OUT 256
#define SIGC    5.0f

// Tiling: 64 batches x 64 outs per workgroup, K staged in chunks of 64.
#define TILE_B 64
#define TILE_O 64
#define KC     64
#define KP     (KC / 2)        // k-pairs per stage
#define NSTAGE (IN_DIM / KC)   // 8

// ---------------------------------------------------------------------------
// CDNA5 async global->LDS path selection.
//  1) clang builtins (present in this toolchain; params are generic v4i*)
//  2) inline asm per CDNA5 ISA (VGLOBAL GLOBAL_LOAD_ASYNC_TO_LDS_B128)
//  3) plain copy (host-compile parse only; never codegen'd for gfx1250)
// ---------------------------------------------------------------------------
#if defined(__HIP_DEVICE_COMPILE__) && defined(__gfx1250__)
  #define ON_GFX1250 1
#else
  #define ON_GFX1250 0
#endif

#if defined(__HIP_DEVICE_COMPILE__) && __has_builtin(__builtin_amdgcn_global_load_async_to_lds_b128)
  #define ASYNC_LOAD_BUILTIN 1
#else
  #define ASYNC_LOAD_BUILTIN 0
#endif

typedef __attribute__((__vector_size__(4 * sizeof(int)))) int v4i;

__device__ __forceinline__ void g2l_b128(const void* g, void* l) {
#if ASYNC_LOAD_BUILTIN
  // Toolchain signature (from diagnostics): generic v4i* params, imm offset,
  // imm cpol. Global source first, LDS destination second.
  __builtin_amdgcn_global_load_async_to_lds_b128((v4i*)g, (v4i*)l, 0, 0);
#elif ON_GFX1250
  unsigned lds_off = (unsigned)(unsigned long long)l;
  asm volatile("global_load_async_to_lds_b128 %0, %1, off"
               :: "v"(lds_off), "v"(g) : "memory");
#else
  *(float4*)l = *(const float4*)g;   // host-parse fallback only
#endif
}

#if defined(__HIP_DEVICE_COMPILE__) && __has_builtin(__builtin_amdgcn_s_wait_asynccnt)
  #define ASYNC_WAIT(N) __builtin_amdgcn_s_wait_asynccnt(N)
#elif ON_GFX1250
  #define ASYNC_WAIT(N) asm volatile("s_wait_asynccnt " #N ::: "memory")
#else
  #define ASYNC_WAIT(N) do {} while (0)
#endif

// ---------------------------------------------------------------------------
// Main kernel: S[b,o] = prod_k ( a[o,k] + w[o,k] * x[b,k] ),  a = 1 - w.
// 256 threads = 8 wave32; 16x16 logical thread grid; 4x4 register tile each.
// Double-buffered async staging of (w,a) tiles and x tiles into LDS.
// LDS element granularity = 2 k-steps, so each DS read feeds 2 FMAs:
//   sWA element = float4 (w0,a0,w1,a1); sX element = float2 (x0,x1).
// ---------------------------------------------------------------------------
template <bool HAS_WS>
__global__ __launch_bounds__(256)
void logic_prod_kernel(const float*  __restrict__ x,
                       const float*  __restrict__ Wraw,
                       const float2* __restrict__ WA,
                       float*        __restrict__ out) {
  // Row strides: sWA row = 33*16B = 528B (== 4 banks mod 64, 16B aligned),
  //              sX  row = 34*8B  = 272B (== 4 banks mod 64, 16B aligned).
  // Both conflict-free for the read patterns below; async b128 dsts aligned.
  __shared__ __align__(16) float4 sWA[2][TILE_O][KP + 1];  // 2*64*33*16 = 67,584 B
  __shared__ __align__(16) float2 sX [2][TILE_B][KP + 2];  // 2*64*34*8  = 34,816 B

  const int tid   = threadIdx.x;        // 0..255
  const int tx    = tid & 15;           // out-tile lane
  const int ty    = tid >> 4;           // batch-tile lane
  const int bBase = blockIdx.x * TILE_B;
  const int oBase = blockIdx.y * TILE_O;

  // Stage one KC-chunk of (w,a) and x tiles into LDS buffer `buf`.
  auto stage = [&](int buf, int k0) {
    // (w,a) tile: 64 rows x 32 pairs x 16B = 32 KB = 2048 b128 -> 8/thread.
#pragma unroll
    for (int i = 0; i < 8; ++i) {
      int t = tid + i * 256;
      int o = t >> 5;          // 0..63
      int c = t & 31;          // 16B chunk within row (one float4 = 2 k's)
      if constexpr (HAS_WS) {
        g2l_b128(&WA[(size_t)(oBase + o) * IN_DIM + k0 + c * 2],
                 &sWA[buf][o][c]);
      } else {
        const float* wr = &Wraw[(size_t)(oBase + o) * IN_DIM + k0 + c * 2];
        float r0 = wr[0], r1 = wr[1];
        float w0 = 1.0f / (1.0f + expf(-SIGC * r0));
        float w1 = 1.0f / (1.0f + expf(-SIGC * r1));
        sWA[buf][o][c] = make_float4(w0, 1.0f - w0, w1, 1.0f - w1);
      }
    }
    // x tile: 64 rows x 64 k x 4B = 16 KB = 1024 b128 -> 4/thread.
#pragma unroll
    for (int i = 0; i < 4; ++i) {
      int t = tid + i * 256;
      int b = t >> 4;          // 0..63
      int c = t & 15;          // 16B chunk (4 k's = 2 float2 elements)
      g2l_b128(&x[(size_t)(bBase + b) * IN_DIM + k0 + c * 4],
               &sX[buf][b][c * 2]);
    }
  };

  float acc[4][4];
#pragma unroll
  for (int i = 0; i < 4; ++i)
#pragma unroll
    for (int j = 0; j < 4; ++j) acc[i][j] = 1.0f;

  stage(0, 0);  // prologue: fill buffer 0

  for (int s = 0; s < NSTAGE; ++s) {
    const int buf = s & 1;
    if (s + 1 < NSTAGE) {
      stage(buf ^ 1, (s + 1) * KC);          // overlap next stage's copies
      // Exactly 12 (HAS_WS) / 4 (!HAS_WS) async issues per wave per stage;
      // waiting to that count retires all of stage s, leaving s+1 in flight.
      if constexpr (HAS_WS) { ASYNC_WAIT(12); } else { ASYNC_WAIT(4); }
    } else {
      ASYNC_WAIT(0);
    }
    __syncthreads();   // stage-s data from all waves now visible

#pragma unroll 2
    for (int p = 0; p < KP; ++p) {        // each p covers two k-steps
      float4 wa[4];
      float2 xv[4];
#pragma unroll
      for (int i = 0; i < 4; ++i) wa[i] = sWA[buf][tx + 16 * i][p];
#pragma unroll
      for (int j = 0; j < 4; ++j) xv[j] = sX[buf][ty + 16 * j][p];
#pragma unroll
      for (int i = 0; i < 4; ++i)
#pragma unroll
        for (int j = 0; j < 4; ++j) {
          acc[i][j] *= fmaf(wa[i].x, xv[j].x, wa[i].y);  // (1-w) + w*x, k
          acc[i][j] *= fmaf(wa[i].z, xv[j].y, wa[i].w);  // (1-w) + w*x, k+1
        }
    }
    __syncthreads();   // everyone done reading buf before it is re-filled
  }

  // out is [BATCH, NUM_OUT] row-major; lanes vary tx -> coalesced stores.
#pragma unroll
  for (int j = 0; j < 4; ++j)
#pragma unroll
    for (int i = 0; i < 4; ++i)
      out[(size_t)(bBase + ty + 16 * j) * NUM_OUT + (oBase + tx + 16 * i)] =
          acc[i][j];
}

// Explicit instantiations so the main kernel is emitted first in the object
// (lets the disasm snippet show the async global->LDS prologue).
template __global__ void logic_prod_kernel<true>(const float* __restrict__,
                                                 const float* __restrict__,
                                                 const float2* __restrict__,
                                                 float* __restrict__);
template __global__ void logic_prod_kernel<false>(const float* __restrict__,
                                                  const float* __restrict__,
                                                  const float2* __restrict__,
                                                  float* __restrict__);

// ---------------------------------------------------------------------------
// Precompute (w, 1-w) pairs with w = sigmoid(5 * W_raw): removes all
// transcendentals from the hot loop; 1 MB table in workspace.
// ---------------------------------------------------------------------------
__global__ void sigmoid_prep_kernel(const float* __restrict__ Wraw,
                                    float2* __restrict__ WA, int n) {
  int i = blockIdx.x * blockDim.x + threadIdx.x;
  if (i < n) {
    float w = 1.0f / (1.0f + expf(-SIGC * Wraw[i]));
    WA[i] = make_float2(w, 1.0f - w);
  }
}

// ---------------------------------------------------------------------------
extern "C" void kernel_launch(void* const* d_in, const int* in_sizes, int n_in,
                              void* d_out, int out_size, void* d_ws,
                              size_t ws_size, hipStream_t stream) {
  (void)in_sizes; (void)n_in; (void)out_size;
  const float* x    = (const float*)d_in[0];   // [2048, 512] f32
  const float* Wraw = (const float*)d_in[1];   // [256, 512]  f32
  float*       out  = (float*)d_out;           // [2048, 256] f32

  const size_t waBytes = (size_t)NUM_OUT * IN_DIM * sizeof(float2);  // 1 MB
  dim3 grid(BATCH / TILE_B, NUM_OUT / TILE_O);                       // 32 x 4

  if (ws_size >= waBytes) {
    float2* WA = (float2*)d_ws;
    int n = NUM_OUT * IN_DIM;
    sigmoid_prep_kernel<<<(n + 255) / 256, 256, 0, stream>>>(Wraw, WA, n);
    logic_prod_kernel<true><<<grid, 256, 0, stream>>>(x, Wraw, WA, out);
  } else {
    logic_prod_kernel<false><<<grid, 256, 0, stream>>>(x, Wraw, nullptr, out);
  }
}